// HashEncoderWithTracking_32272384262741
// MI455X (gfx1250) — compile-verified
//
#include <hip/hip_runtime.h>
#include <stdint.h>
#include <stddef.h>

// 2-wide float vector (avoid HIP float2 struct so nontemporal builtins work)
typedef float v2f __attribute__((ext_vector_type(2)));
// matches the builtin's parameter type: int __vector(4)
typedef int v4i __attribute__((vector_size(16)));

#define AS1 __attribute__((address_space(1)))
#define AS3 __attribute__((address_space(3)))

#if defined(__HIP_DEVICE_COMPILE__)
#if __has_builtin(__builtin_amdgcn_global_load_async_to_lds_b128)
#define HASH_ASYNC_STAGE 1
#endif
#endif

#define NPTS 256   // points per block (8 wave32 waves)
#define NLVL 16
#define PAD  17    // 16 v2f per point, padded to 17 (odd) -> conflict-free LDS

__global__ __launch_bounds__(NPTS)
void hashenc_ngp_kernel(const float* __restrict__ in,     // (B,3) f32
                        const v2f*   __restrict__ emb,    // (7114752,2) f32
                        float*       __restrict__ out,    // (B,32) f32
                        int B)
{
    const int t   = threadIdx.x;
    const int blk = blockIdx.x;
    const int p   = blk * NPTS + t;

    __shared__ float s_in[NPTS * 3];        // 3 KB input slab
    __shared__ v2f   s_out[NPTS * PAD];     // 34 KB transpose tile

    const bool full = (blk + 1) * NPTS <= B;

    float x = 0.f, y = 0.f, z = 0.f;

#if defined(HASH_ASYNC_STAGE)
    if (full) {
        // Async-copy the block's 256*12 = 3072B of coordinates into LDS.
        // 192 lanes x 16B (B128). CDNA5 async path: tracked by ASYNCcnt.
        if (t < 192) {
            const float* g = in + (size_t)blk * (NPTS * 3) + t * 4;
            __builtin_amdgcn_global_load_async_to_lds_b128(
                (AS1 v4i*)(uintptr_t)g,
                (AS3 v4i*)(uint32_t)(uintptr_t)&s_in[t * 4],
                /*imm_offset=*/0, /*cpol=*/0);
        }
#if __has_builtin(__builtin_amdgcn_s_wait_asynccnt)
        __builtin_amdgcn_s_wait_asynccnt(0);
#else
        asm volatile("s_wait_asynccnt 0" ::: "memory");
#endif
        __syncthreads();
        x = s_in[t * 3 + 0];
        y = s_in[t * 3 + 1];
        z = s_in[t * 3 + 2];
    } else
#endif
    {
        if (p < B) {
            const float* q = in + (size_t)p * 3;
            __builtin_prefetch(q, 0, 3);            // global_prefetch_b8
            x = __builtin_nontemporal_load(q + 0);  // input is read once: NT
            y = __builtin_nontemporal_load(q + 1);
            z = __builtin_nontemporal_load(q + 2);
        }
    }

    // ---- 16 levels, fully unrolled so every constant folds ----
#pragma unroll
    for (int l = 0; l < NLVL; ++l) {
        const unsigned R    = 16u << l;              // resolution per dim
        const float    scl  = (float)R - 1.0f;       // scale = 16*2^l - 1 (exact)
        const int      sh   = 4 + l;                 // log2(R)
        unsigned mask, off;
        if (l < 3) {                                 // dense levels (R^3 <= 2^19)
            mask = (R * R * R) - 1u;                 // 2^12-1 / 2^15-1 / 2^18-1
            off  = (l == 0) ? 0u : ((l == 1) ? 4096u : 36864u);
        } else {                                     // hashed levels
            mask = (1u << 19) - 1u;                  // hsize = 2^19
            off  = 299008u + (unsigned)(l - 3) * 524288u;
        }

        const float px = x * scl + 0.5f;
        const float py = y * scl + 0.5f;
        const float pz = z * scl + 0.5f;
        const float gxf = floorf(px), gyf = floorf(py), gzf = floorf(pz);
        const float fx = px - gxf, fy = py - gyf, fz = pz - gzf;
        const unsigned gx = (unsigned)gxf, gy = (unsigned)gyf, gz = (unsigned)gzf;

        const float wx[2] = {1.0f - fx, fx};
        const float wy[2] = {1.0f - fy, fy};
        const float wz[2] = {1.0f - fz, fz};

        float a0 = 0.0f, a1 = 0.0f;
#pragma unroll
        for (int k = 0; k < 8; ++k) {
            const unsigned cx = gx + (k & 1);
            const unsigned cy = gy + ((k >> 1) & 1);
            const unsigned cz = gz + ((k >> 2) & 1);
            unsigned h;
            if (l < 3) {
                h = (cx + (cy << sh) + (cz << (2 * sh))) & mask;   // dense, wraps edge
            } else {
                h = (cx ^ (cy * 2654435761u) ^ (cz * 805459861u)) & mask;
            }
            const v2f e = emb[(size_t)(h + off)];    // 8B gather, L2-resident (RT)
            const float w = wx[k & 1] * wy[(k >> 1) & 1] * wz[(k >> 2) & 1];
            a0 = fmaf(w, e.x, a0);
            a1 = fmaf(w, e.y, a1);
        }
        v2f r; r.x = a0; r.y = a1;
        s_out[t * PAD + l] = r;                      // stage for coalesced store
    }

    // ---- coalesced NT write-out: 256 pts x 16 v2f = 32 KB per block ----
    __syncthreads();
    v2f* o2 = (v2f*)out;
    const size_t tile2 = (size_t)blk * (NPTS * NLVL);  // v2f units
    const size_t lim2  = (size_t)B * NLVL;
#pragma unroll
    for (int i = 0; i < NLVL; ++i) {
        const unsigned g2 = (unsigned)(i * NPTS + t);  // linear v2f idx in tile
        const size_t  gi  = tile2 + g2;
        if (gi < lim2) {
            const unsigned pt = g2 >> 4, lv = g2 & 15;
            // NT store: don't let the 128MB output stream evict the 57MB
            // embedding table from the 192MB L2.
            __builtin_nontemporal_store(s_out[pt * PAD + lv], &o2[gi]);
        }
    }
}

extern "C" void kernel_launch(void* const* d_in, const int* in_sizes, int n_in,
                              void* d_out, int out_size, void* d_ws, size_t ws_size,
                              hipStream_t stream) {
    const float* in  = (const float*)d_in[0];   // (B,3) f32
    const v2f*   emb = (const v2f*)d_in[1];     // (7114752,2) f32
    float*       out = (float*)d_out;           // (B,32) f32
    const int B = in_sizes[0] / 3;
    const int blocks = (B + NPTS - 1) / NPTS;
    hashenc_ngp_kernel<<<blocks, NPTS, 0, stream>>>(in, emb, out, B);
}